// BaseGCN_59657095741992
// MI455X (gfx1250) — compile-verified
//
#include <hip/hip_runtime.h>
#include <hip/hip_bf16.h>

// ---- problem constants (match reference) ----
#define NN 100000
#define EE 1600000
#define GG 64

typedef __attribute__((ext_vector_type(2))) float v2f;
typedef __attribute__((ext_vector_type(8))) float v8f;

// ------------------------------------------------------------------
// utility: zero a float buffer
// ------------------------------------------------------------------
__global__ void k_zero(float* __restrict__ p, long long n) {
    long long i = (long long)blockIdx.x * blockDim.x + threadIdx.x;
    if (i < n) p[i] = 0.0f;
}

// ------------------------------------------------------------------
// degree: deg[dst] += 1 per edge (self-loop +1 folded into dinv pass)
// ------------------------------------------------------------------
__global__ void k_degree(const int* __restrict__ edst, float* __restrict__ deg, int E) {
    int e = blockIdx.x * blockDim.x + threadIdx.x;
    if (e < E) atomicAdd(&deg[edst[e]], 1.0f);
}

__global__ void k_dinv(float* __restrict__ deg, int n) {
    int i = blockIdx.x * blockDim.x + threadIdx.x;
    if (i < n) {
        float d = deg[i] + 1.0f;           // + self loop; always >= 1
        deg[i] = rsqrtf(d);                // in-place: deg becomes dinv
    }
}

// ------------------------------------------------------------------
// WMMA GEMM: pre[N,Fout] = prelu(Hin,[slope]) @ W
// One wave per 16x16 output tile, K-loop in steps of 4 using
// V_WMMA_F32_16X16X4_F32 (exact fp32, matches reference numerics).
// A layout (ISA 7.12.2, 32-bit 16x4): lanes 0-15 K=0,1 ; lanes 16-31 K=2,3
// D layout: VGPR r -> M=r (lanes 0-15) / M=r+8 (lanes 16-31), N=lane&15
// ------------------------------------------------------------------
__global__ void k_gemm_wmma(const float* __restrict__ Hin,
                            const float* __restrict__ W,
                            float* __restrict__ pre,
                            int Fin, int Fout,
                            const float* __restrict__ slope_ptr,
                            int tilesN, int totalTiles) {
    const int lane = threadIdx.x & 31;
    const int wave = threadIdx.x >> 5;
    const int tile = blockIdx.x * (blockDim.x >> 5) + wave;
    if (tile >= totalTiles) return;               // wave-uniform: EXEC stays all-1s

    const int tM = tile / tilesN;
    const int tN = tile - tM * tilesN;
    const int hl   = lane & 15;
    const int kGrp = lane >> 4;                   // 0 or 1

    const long long row = (long long)tM * 16 + hl;     // A row for this lane
    const int       col = tN * 16 + hl;                // B/D column for this lane
    const float slope = slope_ptr ? slope_ptr[0] : 1.0f;

    v8f acc = {};
    for (int k0 = 0; k0 < Fin; k0 += 4) {
        const int ka = k0 + kGrp * 2;
        float a0 = Hin[row * Fin + ka];
        float a1 = Hin[row * Fin + ka + 1];
        a0 = (a0 >= 0.0f) ? a0 : slope * a0;      // fused PReLU on input
        a1 = (a1 >= 0.0f) ? a1 : slope * a1;
        v2f A; A.x = a0; A.y = a1;
        v2f B; B.x = W[(long long)ka * Fout + col];
               B.y = W[(long long)(ka + 1) * Fout + col];
        acc = __builtin_amdgcn_wmma_f32_16x16x4_f32(
                  false, A, false, B, (short)0, acc, false, false);
    }

#pragma unroll
    for (int r = 0; r < 8; ++r) {
        long long mm = (long long)tM * 16 + r + kGrp * 8;
        pre[mm * Fout + col] = acc[r];
    }
}

// ------------------------------------------------------------------
// aggregation init: out[n,f] = pre[n,f] * dinv[n]^2 (self loop) + b[f]
// ------------------------------------------------------------------
__global__ void k_agg_init(const float* __restrict__ pre,
                           const float* __restrict__ dinv,
                           const float* __restrict__ bias,
                           float* __restrict__ out, int F) {
    long long idx = (long long)blockIdx.x * blockDim.x + threadIdx.x;
    long long total = (long long)NN * F;
    if (idx >= total) return;
    int n = (int)(idx / F);
    int f = (int)(idx - (long long)n * F);
    float di = dinv[n];
    out[idx] = pre[idx] * di * di + bias[f];
}

// ------------------------------------------------------------------
// edge aggregation: out[dst] += pre[src] * dinv[src]*dinv[dst]
// one thread per (edge, 4-feature group): float4 gather + 4 f32 atomics
// ------------------------------------------------------------------
__global__ void k_agg_edges(const float* __restrict__ pre,
                            const int* __restrict__ esrc,
                            const int* __restrict__ edst,
                            const float* __restrict__ dinv,
                            float* __restrict__ out, int F) {
    const int fG = F >> 2;
    long long idx = (long long)blockIdx.x * blockDim.x + threadIdx.x;
    long long total = (long long)EE * fG;
    if (idx >= total) return;
    int e  = (int)(idx / fG);
    int fg = (int)(idx - (long long)e * fG);
    int s = esrc[e], d = edst[e];
    float nrm = dinv[s] * dinv[d];
    const float4 v = *(const float4*)(pre + (long long)s * F + fg * 4);
    float* o = out + (long long)d * F + fg * 4;
    atomicAdd(o + 0, v.x * nrm);
    atomicAdd(o + 1, v.y * nrm);
    atomicAdd(o + 2, v.z * nrm);
    atomicAdd(o + 3, v.w * nrm);
}

// ------------------------------------------------------------------
// global mean pool (sums + counts); F = 128 fixed
// ------------------------------------------------------------------
__global__ void k_pool(const float* __restrict__ h,
                       const int* __restrict__ batch,
                       float* __restrict__ pooled,
                       float* __restrict__ cnt) {
    long long idx = (long long)blockIdx.x * blockDim.x + threadIdx.x;
    long long total = (long long)NN * 32;     // 128/4 groups
    if (idx >= total) return;
    int n  = (int)(idx >> 5);
    int fg = (int)(idx & 31);
    int g = batch[n];
    const float4 v = *(const float4*)(h + (long long)n * 128 + fg * 4);
    float* p = pooled + (long long)g * 128 + fg * 4;
    atomicAdd(p + 0, v.x);
    atomicAdd(p + 1, v.y);
    atomicAdd(p + 2, v.z);
    atomicAdd(p + 3, v.w);
    if (fg == 0) atomicAdd(&cnt[g], 1.0f);
}

// ------------------------------------------------------------------
// final: out[g,c] = (pooled[g]/cnt[g]) @ Wlin + blin   (64x128x4, tiny)
// ------------------------------------------------------------------
__global__ void k_final(const float* __restrict__ pooled,
                        const float* __restrict__ cnt,
                        const float* __restrict__ Wlin,
                        const float* __restrict__ blin,
                        float* __restrict__ out) {
    int idx = threadIdx.x;            // 256 threads = G*4
    if (idx >= GG * 4) return;
    int g = idx >> 2, c = idx & 3;
    float cn = fmaxf(cnt[g], 1.0f);
    float acc = 0.0f;
    const float* p = pooled + (long long)g * 128;
    for (int f = 0; f < 128; ++f) acc += p[f] * Wlin[f * 4 + c];
    out[idx] = acc / cn + blin[c];
}

// ------------------------------------------------------------------
// host-side orchestration
// ------------------------------------------------------------------
static inline int cdiv_ll(long long a, int b) { return (int)((a + b - 1) / b); }

extern "C" void kernel_launch(void* const* d_in, const int* in_sizes, int n_in,
                              void* d_out, int out_size, void* d_ws, size_t ws_size,
                              hipStream_t stream) {
    (void)in_sizes; (void)n_in; (void)out_size; (void)ws_size;

    const float* x     = (const float*)d_in[0];   // [N,8]
    const int*   esrc  = (const int*)  d_in[1];   // [E]
    const int*   edst  = (const int*)  d_in[2];   // [E]
    const int*   batch = (const int*)  d_in[3];   // [N]
    const float* W1 = (const float*)d_in[4];  const float* b1 = (const float*)d_in[5];
    const float* W2 = (const float*)d_in[6];  const float* b2 = (const float*)d_in[7];
    const float* W3 = (const float*)d_in[8];  const float* b3 = (const float*)d_in[9];
    const float* W4 = (const float*)d_in[10]; const float* b4 = (const float*)d_in[11];
    const float* a1 = (const float*)d_in[12];
    const float* a2 = (const float*)d_in[13];
    const float* a3 = (const float*)d_in[14];
    const float* Wlin = (const float*)d_in[15];
    const float* blin = (const float*)d_in[16];
    float* out = (float*)d_out;

    // workspace carve-up (all 16B-aligned offsets)
    float* ws     = (float*)d_ws;
    float* dinv   = ws;                                   // N      (deg -> dinv in place)
    float* bufA   = dinv + NN;                            // N*128
    float* bufB   = bufA + (size_t)NN * 128;              // N*128
    float* pre    = bufB + (size_t)NN * 128;              // N*128
    float* pooled = pre  + (size_t)NN * 128;              // G*128
    float* cnt    = pooled + (size_t)GG * 128;            // G

    const int B = 256;

    // ---- normalization ----
    k_zero<<<cdiv_ll(NN, B), B, 0, stream>>>(dinv, NN);
    k_zero<<<cdiv_ll(GG * 128 + GG, B), B, 0, stream>>>(pooled, GG * 128 + GG);
    k_degree<<<cdiv_ll(EE, B), B, 0, stream>>>(edst, dinv, EE);
    k_dinv<<<cdiv_ll(NN, B), B, 0, stream>>>(dinv, NN);

    const int tilesM = NN / 16;   // 6250, exact

    // ---- layer 1: x(8) -> 16 ----
    {
        int tilesN = 16 / 16, tot = tilesM * tilesN;
        k_gemm_wmma<<<cdiv_ll(tot, 4), 128, 0, stream>>>(x, W1, pre, 8, 16, nullptr, tilesN, tot);
        k_agg_init<<<cdiv_ll((long long)NN * 16, B), B, 0, stream>>>(pre, dinv, b1, bufA, 16);
        k_agg_edges<<<cdiv_ll((long long)EE * 4, B), B, 0, stream>>>(pre, esrc, edst, dinv, bufA, 16);
    }
    // ---- layer 2: prelu(bufA,a1)(16) -> 32 ----
    {
        int tilesN = 32 / 16, tot = tilesM * tilesN;
        k_gemm_wmma<<<cdiv_ll(tot, 4), 128, 0, stream>>>(bufA, W2, pre, 16, 32, a1, tilesN, tot);
        k_agg_init<<<cdiv_ll((long long)NN * 32, B), B, 0, stream>>>(pre, dinv, b2, bufB, 32);
        k_agg_edges<<<cdiv_ll((long long)EE * 8, B), B, 0, stream>>>(pre, esrc, edst, dinv, bufB, 32);
    }
    // ---- layer 3: prelu(bufB,a2)(32) -> 64 ----
    {
        int tilesN = 64 / 16, tot = tilesM * tilesN;
        k_gemm_wmma<<<cdiv_ll(tot, 4), 128, 0, stream>>>(bufB, W3, pre, 32, 64, a2, tilesN, tot);
        k_agg_init<<<cdiv_ll((long long)NN * 64, B), B, 0, stream>>>(pre, dinv, b3, bufA, 64);
        k_agg_edges<<<cdiv_ll((long long)EE * 16, B), B, 0, stream>>>(pre, esrc, edst, dinv, bufA, 64);
    }
    // ---- layer 4: prelu(bufA,a3)(64) -> 128 (no output PReLU) ----
    {
        int tilesN = 128 / 16, tot = tilesM * tilesN;
        k_gemm_wmma<<<cdiv_ll(tot, 4), 128, 0, stream>>>(bufA, W4, pre, 64, 128, a3, tilesN, tot);
        k_agg_init<<<cdiv_ll((long long)NN * 128, B), B, 0, stream>>>(pre, dinv, b4, bufB, 128);
        k_agg_edges<<<cdiv_ll((long long)EE * 32, B), B, 0, stream>>>(pre, esrc, edst, dinv, bufB, 128);
    }

    // ---- pool + classifier ----
    k_pool<<<cdiv_ll((long long)NN * 32, B), B, 0, stream>>>(bufB, batch, pooled, cnt);
    k_final<<<1, 256, 0, stream>>>(pooled, cnt, Wlin, blin, out);
}